// ToBEVConvolution_58205396796000
// MI455X (gfx1250) — compile-verified
//
#include <hip/hip_runtime.h>
#include <stddef.h>

typedef unsigned int uint32;
typedef float v2f __attribute__((ext_vector_type(2)));
typedef float v8f __attribute__((ext_vector_type(8)));

#define BASE 64
#define KEYSPACE (64 * 64 * 64)   // 262144
#define IN_C 32
#define OUT_C 32
#define NKERN 64
#define GEMM_BLOCKS 1024          // 4 waves/block -> 4096 persistent waves

// ---------------------------------------------------------------------------
// K1: mark key occupancy bitmap + per-kidx histogram (LDS-staged)
// ---------------------------------------------------------------------------
__global__ void markHist(const int* __restrict__ coords,
                         const int* __restrict__ stridep,
                         uint32* __restrict__ rank,     // occupancy flags (KEYSPACE)
                         uint32* __restrict__ hist,     // 64 counters
                         int N) {
  __shared__ uint32 lh[NKERN];
  int t = threadIdx.x;
  if (t < NKERN) lh[t] = 0u;
  __syncthreads();

  int n = blockIdx.x * blockDim.x + t;
  if (n < N) {
    int c0 = coords[n * 4 + 0];
    int c1 = coords[n * 4 + 1];
    int c2 = coords[n * 4 + 2];
    int c3 = coords[n * 4 + 3];
    int s  = *stridep;
    int kidx = (c1 / s) & (NKERN - 1);
    uint32 key = ((uint32)c0 * (uint32)BASE + (uint32)c2) * (uint32)BASE + (uint32)c3;
    rank[key] = 1u;                 // benign race: everyone writes 1
    atomicAdd(&lh[kidx], 1u);
  }
  __syncthreads();
  if (t < NKERN) {
    uint32 v = lh[t];
    if (v) atomicAdd(&hist[t], v);
  }
}

// ---------------------------------------------------------------------------
// K2: tiny serial scans: group starts/cursors + tile starts (64 entries)
// ---------------------------------------------------------------------------
__global__ void scanGroups(const uint32* __restrict__ hist,
                           uint32* __restrict__ gstart,
                           uint32* __restrict__ gcursor,
                           uint32* __restrict__ tstart) {
  if (threadIdx.x == 0 && blockIdx.x == 0) {
    uint32 run = 0, trun = 0;
    for (int k = 0; k < NKERN; ++k) {
      gstart[k] = run;
      gcursor[k] = run;
      run += hist[k];
      tstart[k] = trun;
      trun += (hist[k] + 15u) >> 4;
    }
    tstart[NKERN] = trun;   // total tiles
  }
}

// ---------------------------------------------------------------------------
// K3a: per-256 block exclusive scan of occupancy flags (in place) + block sums
// ---------------------------------------------------------------------------
__global__ void scanBlock256(uint32* __restrict__ rank,
                             uint32* __restrict__ blockSum) {
  __shared__ uint32 tmp[256];
  int t = threadIdx.x;
  int i = blockIdx.x * 256 + t;
  uint32 f = rank[i];
  tmp[t] = f;
  __syncthreads();
  for (int off = 1; off < 256; off <<= 1) {
    uint32 v = (t >= off) ? tmp[t - off] : 0u;
    __syncthreads();
    tmp[t] += v;
    __syncthreads();
  }
  rank[i] = tmp[t] - f;                       // exclusive within block
  if (t == 255) blockSum[blockIdx.x] = tmp[t];
}

// ---------------------------------------------------------------------------
// K3b: exclusive scan of the 1024 block sums (single 1024-thread block)
// ---------------------------------------------------------------------------
__global__ void scanBlock1024(uint32* __restrict__ blockSum) {
  __shared__ uint32 tmp[1024];
  int t = threadIdx.x;
  uint32 f = blockSum[t];
  tmp[t] = f;
  __syncthreads();
  for (int off = 1; off < 1024; off <<= 1) {
    uint32 v = (t >= off) ? tmp[t - off] : 0u;
    __syncthreads();
    tmp[t] += v;
    __syncthreads();
  }
  blockSum[t] = tmp[t] - f;
}

// ---------------------------------------------------------------------------
// K4: counting-sort scatter (group points by kidx) + sorted-unique inverse
// ---------------------------------------------------------------------------
__global__ void scatterInv(const int* __restrict__ coords,
                           const int* __restrict__ stridep,
                           const uint32* __restrict__ rank,
                           const uint32* __restrict__ blockOff,
                           uint32* __restrict__ gcursor,
                           uint32* __restrict__ order,
                           uint32* __restrict__ inv,
                           int N) {
  int n = blockIdx.x * blockDim.x + threadIdx.x;
  if (n >= N) return;
  int c0 = coords[n * 4 + 0];
  int c1 = coords[n * 4 + 1];
  int c2 = coords[n * 4 + 2];
  int c3 = coords[n * 4 + 3];
  int s  = *stridep;
  int kidx = (c1 / s) & (NKERN - 1);
  uint32 key = ((uint32)c0 * (uint32)BASE + (uint32)c2) * (uint32)BASE + (uint32)c3;
  uint32 pos = atomicAdd(&gcursor[kidx], 1u);
  order[pos] = (uint32)n;
  inv[n] = rank[key] + blockOff[key >> 8];    // sorted-unique index of key
}

// ---------------------------------------------------------------------------
// K5: persistent-wave WMMA GEMM. Each wave owns a contiguous chunk of tiles;
//     B fragments (kernel[k]) stay in VGPRs and reload only on group change.
//     D[16x32] = A(feats tile 16x32) x B(kernel[k] 32x32), f32 WMMA 16x16x4.
//     Invalid tile rows are CLAMPED (not masked): D rows >= cnt are simply
//     never stored, so no exec-divergent loads are needed in the hot path.
// ---------------------------------------------------------------------------
__global__ void __launch_bounds__(128)
bevGemm(const float* __restrict__ feats,
        const float* __restrict__ kernelW,
        const uint32* __restrict__ order,
        const uint32* __restrict__ inv,
        const uint32* __restrict__ hist,
        const uint32* __restrict__ gstart,
        const uint32* __restrict__ tstart,
        float* __restrict__ out) {
  int wave = threadIdx.x >> 5;
  int lane = threadIdx.x & 31;
  int waveGid = blockIdx.x * 4 + wave;
  int numWaves = gridDim.x * 4;

  uint32 totalTiles = tstart[NKERN];
  uint32 chunk = (totalTiles + (uint32)numWaves - 1u) / (uint32)numWaves;
  uint32 t0 = (uint32)waveGid * chunk;
  if (t0 >= totalTiles) return;               // wave-uniform
  uint32 t1 = t0 + chunk;
  if (t1 > totalTiles) t1 = totalTiles;

  // lane roles (constant across tiles)
  int r    = lane & 15;            // A row within tile
  int nlo  = lane & 15;            // D column (low 16)
  int koff = (lane >> 4) << 1;     // K sub-offset: 0 (lanes 0-15) or 2 (16-31)
  int hi8  = (lane >> 4) * 8;      // D row offset per lane half

  // binary search group for first tile: largest k with tstart[k] <= t0
  int k = 0;
#pragma unroll
  for (int s = 32; s > 0; s >>= 1)
    if (k + s < NKERN && tstart[k + s] <= t0) k += s;

  int kCur = -1;
  uint32 gstartK = 0, histK = 0, tstartK = 0;
  v2f b0[8], b1[8];

  for (uint32 t = t0; t < t1; ++t) {
    // advance group (wave-uniform; groups with zero tiles are skipped)
    while (k + 1 < NKERN && tstart[k + 1] <= t) ++k;

    if (k != kCur) {                          // reload B fragments (rare)
      kCur = k;
      gstartK = gstart[k];
      histK   = hist[k];
      tstartK = tstart[k];
      const float* Bk = kernelW + (size_t)k * (IN_C * OUT_C);
#pragma unroll
      for (int kk = 0; kk < 8; ++kk) {
        int kb = 4 * kk + koff;
        b0[kk].x = Bk[kb * OUT_C + nlo];
        b0[kk].y = Bk[(kb + 1) * OUT_C + nlo];
        b1[kk].x = Bk[kb * OUT_C + nlo + 16];
        b1[kk].y = Bk[(kb + 1) * OUT_C + nlo + 16];
      }
    }

    uint32 tig  = t - tstartK;
    uint32 base = gstartK + tig * 16u;
    uint32 rem  = histK - tig * 16u;
    int cnt = (rem > 16u) ? 16 : (int)rem;    // >= 1 for any existing tile

    // prefetch next tile's gather indices (global_prefetch_b8)
    __builtin_prefetch(&order[base + 16u + (uint32)r], 0, 0);

    // clamped row: invalid rows duplicate the last valid row (results unused)
    int rc = (r < cnt) ? r : (cnt - 1);
    uint32 pidx  = order[base + (uint32)rc];
    uint32 myInv = inv[pidx];
    const float* frow = feats + (size_t)pidx * IN_C;

    v2f a[8];
#pragma unroll
    for (int kk = 0; kk < 8; ++kk) {
      a[kk] = *(const v2f*)(frow + 4 * kk + koff);  // global_load_b64, clause-able
    }

    v8f acc0, acc1;
#pragma unroll
    for (int j = 0; j < 8; ++j) { acc0[j] = 0.f; acc1[j] = 0.f; }

#pragma unroll
    for (int kk = 0; kk < 8; ++kk) {
      acc0 = __builtin_amdgcn_wmma_f32_16x16x4_f32(false, a[kk], false, b0[kk],
                                                   (short)0, acc0, false, false);
      acc1 = __builtin_amdgcn_wmma_f32_16x16x4_f32(false, a[kk], false, b1[kk],
                                                   (short)0, acc1, false, false);
    }

    // D layout: VGPR j -> row j (lanes 0-15) / row j+8 (lanes 16-31), col = lane&15
#pragma unroll
    for (int j = 0; j < 8; ++j) {
      int m = j + hi8;
      uint32 rowInv = (uint32)__shfl((int)myInv, m, 32);  // inv of tile-row m
      if (m < cnt) {
        float* o = out + (size_t)rowInv * OUT_C;
        __hip_atomic_fetch_add(&o[nlo], acc0[j], __ATOMIC_RELAXED,
                               __HIP_MEMORY_SCOPE_AGENT);
        __hip_atomic_fetch_add(&o[nlo + 16], acc1[j], __ATOMIC_RELAXED,
                               __HIP_MEMORY_SCOPE_AGENT);
      }
    }
  }
}

// ---------------------------------------------------------------------------
// launch
// ---------------------------------------------------------------------------
extern "C" void kernel_launch(void* const* d_in, const int* in_sizes, int n_in,
                              void* d_out, int out_size, void* d_ws, size_t ws_size,
                              hipStream_t stream) {
  const int*   coords  = (const int*)d_in[0];
  const float* feats   = (const float*)d_in[1];
  const float* kernelW = (const float*)d_in[2];
  const int*   stridep = (const int*)d_in[3];
  int N = in_sizes[0] / 4;

  // workspace layout (uint32 units)
  uint32* ws       = (uint32*)d_ws;
  uint32* rank     = ws;                       // KEYSPACE
  uint32* blockOff = rank + KEYSPACE;          // 1024
  uint32* hist     = blockOff + 1024;          // 64
  uint32* gstart   = hist + NKERN;             // 64
  uint32* gcursor  = gstart + NKERN;           // 64
  uint32* tstart   = gcursor + NKERN;          // 65 (+pad)
  uint32* order    = tstart + 72;              // N (+32 pad for prefetch)
  uint32* inv      = order + N + 32;           // N

  // zero output (padded rows must be 0) and occupancy+hist
  hipMemsetAsync(d_out, 0, (size_t)out_size * sizeof(float), stream);
  hipMemsetAsync(ws, 0, (size_t)(KEYSPACE + 1024 + NKERN) * sizeof(uint32), stream);

  int pblocks = (N + 255) / 256;
  markHist<<<pblocks, 256, 0, stream>>>(coords, stridep, rank, hist, N);
  scanGroups<<<1, 64, 0, stream>>>(hist, gstart, gcursor, tstart);
  scanBlock256<<<KEYSPACE / 256, 256, 0, stream>>>(rank, blockOff);
  scanBlock1024<<<1, 1024, 0, stream>>>(blockOff);
  scatterInv<<<pblocks, 256, 0, stream>>>(coords, stridep, rank, blockOff,
                                          gcursor, order, inv, N);

  bevGemm<<<GEMM_BLOCKS, 128, 0, stream>>>(feats, kernelW, order, inv,
                                           hist, gstart, tstart, (float*)d_out);
}